// GNNNodeModule_87617332838899
// MI455X (gfx1250) — compile-verified
//
#include <hip/hip_runtime.h>

typedef __attribute__((ext_vector_type(2))) float v2f;
typedef __attribute__((ext_vector_type(4))) float v4f;
typedef __attribute__((ext_vector_type(8))) float v8f;

#define DF 64

__device__ __forceinline__ v8f wmma_f32_k4(v2f a, v2f b, v8f c) {
  // D = A(16x4 f32) * B(4x16 f32) + C(16x16 f32)
  return __builtin_amdgcn_wmma_f32_16x16x4_f32(false, a, false, b, (short)0, c,
                                               false, false);
}

// ---------------------------------------------------------------------------
// Edge update: edges = MLP([edges | nodes[snd] | nodes[rcv] | g]) (K=256)
// Also: received[rcv] += edges_new (f32 atomics), per-block sum of edges_new.
// One wave per 16-edge tile, 2 waves per block. Edge stream is non-temporal
// (410MB/step, zero reuse) so WGP$/L2 stay hot for weights + node gathers.
// ---------------------------------------------------------------------------
__global__ __launch_bounds__(64) void gnn_edge_kernel(
    const float* __restrict__ nodes,
    float* __restrict__ edges,            // updated in place
    const float* __restrict__ g,
    const int* __restrict__ senders,
    const int* __restrict__ receivers,
    const float* __restrict__ W1, const float* __restrict__ b1,
    const float* __restrict__ W2, const float* __restrict__ b2,
    float* __restrict__ received,
    float* __restrict__ agg_part,
    int numTiles)
{
  constexpr int XS = 260;                 // 256 + pad; 1040B rows: 16B aligned
  __shared__ __align__(16) float xs[2][16][XS];   // 33,280 B
  __shared__ int   ridx[2][16];
  __shared__ float red[2][4][32];

  const int wave = threadIdx.x >> 5;
  const int lane = threadIdx.x & 31;
  const int tile = blockIdx.x * 2 + wave;
  const bool active = tile < numTiles;    // wave-uniform
  const int t = active ? tile : 0;
  const long ebase = (long)t * (16 * DF);

  // ---- stage x = [edge | sent | recv | g] : 16 rows x 256 cols ----
  for (int i = lane; i < 256; i += 32) {
    const int r = i >> 4, q = (i & 15) << 2;
    v4f v = __builtin_nontemporal_load(
        (const v4f*)(edges + ebase + r * DF + q));   // streaming, TH=NT
    *(v4f*)&xs[wave][r][q] = v;
  }
  for (int i = lane; i < 256; i += 32) {
    const int r = i >> 4, q = (i & 15) << 2;
    const int si = senders[t * 16 + r];
    v4f v = *(const v4f*)(nodes + (long)si * DF + q);
    *(v4f*)&xs[wave][r][64 + q] = v;
  }
  for (int i = lane; i < 256; i += 32) {
    const int r = i >> 4, q = (i & 15) << 2;
    const int ri = receivers[t * 16 + r];
    if (q == 0) ridx[wave][r] = ri;
    v4f v = *(const v4f*)(nodes + (long)ri * DF + q);
    *(v4f*)&xs[wave][r][128 + q] = v;
  }
  for (int i = lane; i < 256; i += 32) {
    const int r = i >> 4, q = (i & 15) << 2;
    v4f v = *(const v4f*)(g + q);
    *(v4f*)&xs[wave][r][192 + q] = v;
  }
  __syncthreads();

  const int h2 = lane >> 4;               // K-half selector (ISA A/B layout)
  const int ln = lane & 15;               // M (for A) / N (for B,C)

  // ---- layer 1: [16x256] @ [256x64] -> relu ----
  v8f acc[4] = {};
  for (int kb = 0; kb < 64; ++kb) {
    const int kc = 4 * kb + 2 * h2;
    v2f a = *(const v2f*)&xs[wave][ln][kc];
    const float* wp = W1 + kc * DF + ln;
#pragma unroll
    for (int nt = 0; nt < 4; ++nt) {
      v2f b = { wp[nt * 16], wp[DF + nt * 16] };
      acc[nt] = wmma_f32_k4(a, b, acc[nt]);
    }
  }

  // relu + bias, stash h (16x64) back into LDS rows
#pragma unroll
  for (int nt = 0; nt < 4; ++nt) {
    const float bb = b1[nt * 16 + ln];
#pragma unroll
    for (int r = 0; r < 8; ++r) {
      float v = acc[nt][r] + bb;
      v = v > 0.f ? v : 0.f;
      xs[wave][r + 8 * h2][nt * 16 + ln] = v;   // wave-local; DS is in-order
    }
  }

  // ---- layer 2: [16x64] @ [64x64] ----
  v8f o[4] = {};
  for (int kb = 0; kb < 16; ++kb) {
    const int kc = 4 * kb + 2 * h2;
    v2f a = *(const v2f*)&xs[wave][ln][kc];
    const float* wp = W2 + kc * DF + ln;
#pragma unroll
    for (int nt = 0; nt < 4; ++nt) {
      v2f b = { wp[nt * 16], wp[DF + nt * 16] };
      o[nt] = wmma_f32_k4(a, b, o[nt]);
    }
  }

  // ---- bias, write back (NT), scatter-add to received, partial sums ----
  float es[4] = {0.f, 0.f, 0.f, 0.f};
#pragma unroll
  for (int nt = 0; nt < 4; ++nt) {
    const float bb = b2[nt * 16 + ln];
#pragma unroll
    for (int r = 0; r < 8; ++r) {
      const int row = r + 8 * h2;
      const float v = o[nt][r] + bb;
      if (active) {
        __builtin_nontemporal_store(v, edges + ebase + row * DF + nt * 16 + ln);
        atomicAdd(&received[(long)ridx[wave][row] * DF + nt * 16 + ln], v);
        es[nt] += v;
      }
    }
  }
#pragma unroll
  for (int nt = 0; nt < 4; ++nt) red[wave][nt][lane] = es[nt];
  __syncthreads();
  {
    const int n = threadIdx.x;            // 0..63
    const int nt = n >> 4, nl = n & 15;
    const float s = red[0][nt][nl] + red[0][nt][nl + 16] +
                    red[1][nt][nl] + red[1][nt][nl + 16];
    agg_part[(long)blockIdx.x * DF + n] = s;
  }
}

// ---------------------------------------------------------------------------
// Node update: nodes = MLP([nodes | received | g]) (K=192), in place.
// Node features keep regular temporal hints (re-read next step via gathers).
// ---------------------------------------------------------------------------
__global__ __launch_bounds__(64) void gnn_node_kernel(
    float* __restrict__ nodes,            // updated in place
    const float* __restrict__ received,
    const float* __restrict__ g,
    const float* __restrict__ W1, const float* __restrict__ b1,
    const float* __restrict__ W2, const float* __restrict__ b2,
    float* __restrict__ agg_part,
    int numTiles)
{
  constexpr int XS = 196;                 // 192 + pad; 784B rows: 16B aligned
  __shared__ __align__(16) float xs[2][16][XS];   // 25,088 B
  __shared__ float red[2][4][32];

  const int wave = threadIdx.x >> 5;
  const int lane = threadIdx.x & 31;
  const int tile = blockIdx.x * 2 + wave;
  const bool active = tile < numTiles;
  const int t = active ? tile : 0;
  const long nbase = (long)t * (16 * DF);

  for (int i = lane; i < 256; i += 32) {
    const int r = i >> 4, q = (i & 15) << 2;
    v4f v = *(const v4f*)(nodes + nbase + r * DF + q);
    *(v4f*)&xs[wave][r][q] = v;
  }
  for (int i = lane; i < 256; i += 32) {
    const int r = i >> 4, q = (i & 15) << 2;
    v4f v = *(const v4f*)(received + nbase + r * DF + q);
    *(v4f*)&xs[wave][r][64 + q] = v;
  }
  for (int i = lane; i < 256; i += 32) {
    const int r = i >> 4, q = (i & 15) << 2;
    v4f v = *(const v4f*)(g + q);
    *(v4f*)&xs[wave][r][128 + q] = v;
  }
  __syncthreads();

  const int h2 = lane >> 4;
  const int ln = lane & 15;

  v8f acc[4] = {};
  for (int kb = 0; kb < 48; ++kb) {       // K = 192
    const int kc = 4 * kb + 2 * h2;
    v2f a = *(const v2f*)&xs[wave][ln][kc];
    const float* wp = W1 + kc * DF + ln;
#pragma unroll
    for (int nt = 0; nt < 4; ++nt) {
      v2f b = { wp[nt * 16], wp[DF + nt * 16] };
      acc[nt] = wmma_f32_k4(a, b, acc[nt]);
    }
  }

#pragma unroll
  for (int nt = 0; nt < 4; ++nt) {
    const float bb = b1[nt * 16 + ln];
#pragma unroll
    for (int r = 0; r < 8; ++r) {
      float v = acc[nt][r] + bb;
      v = v > 0.f ? v : 0.f;
      xs[wave][r + 8 * h2][nt * 16 + ln] = v;
    }
  }

  v8f o[4] = {};
  for (int kb = 0; kb < 16; ++kb) {       // K = 64
    const int kc = 4 * kb + 2 * h2;
    v2f a = *(const v2f*)&xs[wave][ln][kc];
    const float* wp = W2 + kc * DF + ln;
#pragma unroll
    for (int nt = 0; nt < 4; ++nt) {
      v2f b = { wp[nt * 16], wp[DF + nt * 16] };
      o[nt] = wmma_f32_k4(a, b, o[nt]);
    }
  }

  float ns[4] = {0.f, 0.f, 0.f, 0.f};
#pragma unroll
  for (int nt = 0; nt < 4; ++nt) {
    const float bb = b2[nt * 16 + ln];
#pragma unroll
    for (int r = 0; r < 8; ++r) {
      const int row = r + 8 * h2;
      const float v = o[nt][r] + bb;
      if (active) {
        nodes[nbase + row * DF + nt * 16 + ln] = v;
        ns[nt] += v;
      }
    }
  }
#pragma unroll
  for (int nt = 0; nt < 4; ++nt) red[wave][nt][lane] = ns[nt];
  __syncthreads();
  {
    const int n = threadIdx.x;
    const int nt = n >> 4, nl = n & 15;
    const float s = red[0][nt][nl] + red[0][nt][nl + 16] +
                    red[1][nt][nl] + red[1][nt][nl + 16];
    agg_part[(long)blockIdx.x * DF + n] = s;
  }
}

// ---------------------------------------------------------------------------
// Global update: reduce per-block partials, g = MLP([agg_n | agg_e | g]).
// One block of 256 threads (4-way slice of the partial reduction).
// ---------------------------------------------------------------------------
__global__ __launch_bounds__(256) void gnn_global_kernel(
    const float* __restrict__ agg_n_part, int nBlocksN,
    const float* __restrict__ agg_e_part, int nBlocksE,
    float* __restrict__ g,                // updated in place (ws)
    const float* __restrict__ W1, const float* __restrict__ b1,
    const float* __restrict__ W2, const float* __restrict__ b2,
    float* __restrict__ g_final)          // d_out tail
{
  const int t = threadIdx.x;
  const int j = t & 63;
  const int s = t >> 6;                   // 0..3
  float an = 0.f, ae = 0.f;
  for (int i = s; i < nBlocksN; i += 4) an += agg_n_part[(long)i * DF + j];
  for (int i = s; i < nBlocksE; i += 4) ae += agg_e_part[(long)i * DF + j];

  __shared__ float rn[4][64], re[4][64];
  __shared__ float x[192], h[64];
  rn[s][j] = an;
  re[s][j] = ae;
  __syncthreads();
  if (t < 64) {
    x[t]       = rn[0][t] + rn[1][t] + rn[2][t] + rn[3][t];
    x[64 + t]  = re[0][t] + re[1][t] + re[2][t] + re[3][t];
    x[128 + t] = g[t];
  }
  __syncthreads();
  if (t < 64) {
    float acc = b1[t];
    for (int k = 0; k < 192; ++k) acc += x[k] * W1[k * DF + t];
    h[t] = acc > 0.f ? acc : 0.f;
  }
  __syncthreads();
  if (t < 64) {
    float o = b2[t];
    for (int k = 0; k < 64; ++k) o += h[k] * W2[k * DF + t];
    g[t] = o;
    g_final[t] = o;
  }
}

// ---------------------------------------------------------------------------
extern "C" void kernel_launch(void* const* d_in, const int* in_sizes, int n_in,
                              void* d_out, int out_size, void* d_ws, size_t ws_size,
                              hipStream_t stream)
{
  (void)n_in; (void)out_size; (void)ws_size;
  const float* nodes_in  = (const float*)d_in[0];
  const float* edges_in  = (const float*)d_in[1];
  const float* g_in      = (const float*)d_in[2];
  const int*   senders   = (const int*)d_in[3];
  const int*   receivers = (const int*)d_in[4];
  const float* We1 = (const float*)d_in[5];
  const float* be1 = (const float*)d_in[6];
  const float* We2 = (const float*)d_in[7];
  const float* be2 = (const float*)d_in[8];
  const float* Wn1 = (const float*)d_in[9];
  const float* bn1 = (const float*)d_in[10];
  const float* Wn2 = (const float*)d_in[11];
  const float* bn2 = (const float*)d_in[12];
  const float* Wg1 = (const float*)d_in[13];
  const float* bg1 = (const float*)d_in[14];
  const float* Wg2 = (const float*)d_in[15];
  const float* bg2 = (const float*)d_in[16];

  const int N = in_sizes[0] / DF;
  const int E = in_sizes[1] / DF;

  // d_out layout: nodes [N*64] | edges [E*64] | g [64] — used as working bufs.
  float* nodes = (float*)d_out;
  float* edges = (float*)d_out + (size_t)N * DF;
  float* gout  = (float*)d_out + (size_t)(N + E) * DF;

  const int eTiles  = (E + 15) / 16;
  const int nTiles  = (N + 15) / 16;
  const int eBlocks = (eTiles + 1) / 2;
  const int nBlocks = (nTiles + 1) / 2;

  // Workspace: received[N*64] | aggE[eBlocks*64] | aggN[nBlocks*64] | g[64]
  char* ws = (char*)d_ws;
  float* received = (float*)ws;  ws += (size_t)N * DF * sizeof(float);
  float* aggE     = (float*)ws;  ws += (size_t)eBlocks * DF * sizeof(float);
  float* aggN     = (float*)ws;  ws += (size_t)nBlocks * DF * sizeof(float);
  float* gcur     = (float*)ws;  ws += DF * sizeof(float);

  (void)hipMemcpyAsync(nodes, nodes_in, (size_t)N * DF * sizeof(float),
                       hipMemcpyDeviceToDevice, stream);
  (void)hipMemcpyAsync(edges, edges_in, (size_t)E * DF * sizeof(float),
                       hipMemcpyDeviceToDevice, stream);
  (void)hipMemcpyAsync(gcur, g_in, DF * sizeof(float),
                       hipMemcpyDeviceToDevice, stream);

  for (int step = 0; step < 3; ++step) {
    (void)hipMemsetAsync(received, 0, (size_t)N * DF * sizeof(float), stream);
    gnn_edge_kernel<<<eBlocks, 64, 0, stream>>>(
        nodes, edges, gcur, senders, receivers,
        We1, be1, We2, be2, received, aggE, eTiles);
    gnn_node_kernel<<<nBlocks, 64, 0, stream>>>(
        nodes, received, gcur, Wn1, bn1, Wn2, bn2, aggN, nTiles);
    gnn_global_kernel<<<1, 256, 0, stream>>>(
        aggN, nBlocks, aggE, eBlocks, gcur, Wg1, bg1, Wg2, bg2, gout);
  }
}